// MorsePotentialCadherin_63101659513298
// MI455X (gfx1250) — compile-verified
//
#include <hip/hip_runtime.h>
#include <hip/hip_bf16.h>
#include <math.h>

// ---------------------------------------------------------------------------
// Morse pair potential with cadherin-based epsilon, N=6144 all-pairs energy.
//   * prepass: fold one-hot celltype algebra into per-particle packed vectors:
//       POS4[i] = (x, y, z, |p|^2)      -> A/B WMMA fragments = one b64 load
//       PAR4[i] = (s, type, rad, cadL)  -> column params     = one b128 load
//   * main: each wave computes 16x16 dr2 tiles with V_WMMA_F32_16X16X4_F32
//     (homogeneous trick: [-2x,-2y,-2z,1] x [x;y;z;|p|^2] + |p_i|^2), then
//     uses the tile as a cutoff screen: box 40 vs cutoff 2 -> ~88% of tiles
//     have no interacting pair and skip all sqrt/exp work.
//   * raw v_sqrt_f32 via __builtin_amdgcn_sqrtf (no IEEE refinement chain).
// ---------------------------------------------------------------------------

#define NPART 6144
#define TILE 16
#define NTILES (NPART / TILE)                   // 384
#define WAVES_PER_BLOCK 8
#define ROW_BLOCKS (NTILES / WAVES_PER_BLOCK)   // 48
#define COL_CHUNKS 16
#define COLS_PER_CHUNK (NTILES / COL_CHUNKS)    // 24

typedef __attribute__((ext_vector_type(2))) float v2f;
typedef __attribute__((ext_vector_type(8))) float v8f;

__device__ constexpr float kAlpha  = 2.8f;
__device__ constexpr float kRC     = 2.0f;
__device__ constexpr float kRO     = 1.7f;
__device__ constexpr float kRC2    = kRC * kRC;          // 4.0
__device__ constexpr float kRO2    = kRO * kRO;          // 2.89
__device__ constexpr float kInvDen = 1.0f / ((kRC2 - kRO2) * (kRC2 - kRO2) * (kRC2 - kRO2));
__device__ constexpr float kEpsMin = 1.0f;
__device__ constexpr float kEpsMax = 5.0f;

// Workspace layout (floats): [0, 4N): POS4   [4N, 8N): PAR4   [8N, 9N): ALIVE
// Total 9*6144*4 = 221,184 bytes.

__global__ void morse_prep_kernel(const float* __restrict__ pos,
                                  const float* __restrict__ ct,
                                  const float* __restrict__ cad,
                                  const float* __restrict__ rad,
                                  float* __restrict__ W,
                                  float* __restrict__ out) {
    int i = blockIdx.x * blockDim.x + threadIdx.x;
    if (i == 0) out[0] = 0.0f;   // zero the accumulator every launch (graph-safe)
    if (i >= NPART) return;

    float x = pos[3 * i + 0];
    float y = pos[3 * i + 1];
    float z = pos[3 * i + 2];
    float c0 = ct[4 * i + 0], c1 = ct[4 * i + 1];
    float c2 = ct[4 * i + 2], c3 = ct[4 * i + 3];
    float s = c0 * cad[5 * i + 0] + c1 * cad[5 * i + 1] +
              c2 * cad[5 * i + 2] + c3 * cad[5 * i + 3];
    float t = c1 * 1.0f + c2 * 2.0f + c3 * 3.0f;           // one-hot -> index
    float alive = ((c0 + c1 + c2 + c3) > 0.0f) ? 1.0f : 0.0f;

    float4* POS4 = (float4*)W;
    float4* PAR4 = (float4*)(W + 4 * NPART);
    float*  AL   = W + 8 * NPART;
    POS4[i] = make_float4(x, y, z, x * x + y * y + z * z);
    PAR4[i] = make_float4(s, t, rad[i], cad[5 * i + 4]);
    AL[i]   = alive;
}

__global__ __launch_bounds__(256) void morse_energy_kernel(const float* __restrict__ W,
                                                           float* __restrict__ out) {
    const float2* __restrict__ POSP = (const float2*)W;            // (x,y)|(z,p2) pairs
    const float4* __restrict__ POS4 = (const float4*)W;
    const float4* __restrict__ PAR4 = (const float4*)(W + 4 * NPART);
    const float*  __restrict__ AL   = W + 8 * NPART;

    const int lane = threadIdx.x & 31;          // wave32
    const int wave = threadIdx.x >> 5;
    const int half = lane >> 4;                 // 0: lanes 0-15, 1: lanes 16-31
    const int l16  = lane & 15;

    const int rowTile = blockIdx.x * WAVES_PER_BLOCK + wave;
    const int rowBase = rowTile * TILE;

    // --- A fragment (16x4 f32): lane<16 holds {K0,K1}=(x,y), lane>=16 {K2,K3}=(z,1).
    //     One b64 load from POS4: low lanes get (x,y), high lanes get (z,p2).
    const int ri = rowBase + l16;
    float2 va = POSP[2 * ri + half];
    v2f a;
    a[0] = -2.0f * va.x;                        // -2x or -2z
    a[1] = half ? 1.0f : (-2.0f * va.y);        // 1  or -2y

    // --- row-side parameters for this lane's 8 accumulator rows
    //     (C/D layout: VGPR v holds row M = v + 8*half, col N = lane&15)
    float rowP2[8], rowS[8], rowT[8], rowRad[8], rowAl[8];
#pragma unroll
    for (int v = 0; v < 8; ++v) {
        const int rm = rowBase + v + 8 * half;
        float4 pr = PAR4[rm];
        rowS[v]   = pr.x;
        rowT[v]   = pr.y;
        rowRad[v] = pr.z;
        rowP2[v]  = POS4[rm].w;
        rowAl[v]  = AL[rm];
    }

    float acc = 0.0f;
    const int tile0 = blockIdx.y * COLS_PER_CHUNK;

    // --- software-pipelined B fragment: one b64 load per tile.
    //     B (4x16): low lanes {K0,K1}=(x_j,y_j), high lanes {K2,K3}=(z_j,p2_j)
    //     which is exactly the POS4 half this lane loads -> zero ALU fixup.
    int j = tile0 * TILE + l16;
    float2 vb = POSP[2 * j + half];

    for (int t = 0; t < COLS_PER_CHUNK; ++t) {
        const int jcur = j;
        v2f b;
        b[0] = vb.x;
        b[1] = vb.y;

        // C init = |p_i|^2 ; WMMA -> dr2 tile
        v8f c;
#pragma unroll
        for (int v = 0; v < 8; ++v) c[v] = rowP2[v];
        c = __builtin_amdgcn_wmma_f32_16x16x4_f32(
                /*neg_a=*/false, a, /*neg_b=*/false, b,
                /*c_mod=*/(short)0, c, /*reuse_a=*/false, /*reuse_b=*/false);

        // prefetch next tile's B fragment while the WMMA/screen completes
        if (t + 1 < COLS_PER_CHUNK) {
            j = jcur + TILE;
            vb = POSP[2 * j + half];
        }

        // --- cutoff screen: skip entire tile if every dr2 >= rc^2
        float mn = fminf(fminf(fminf(c[0], c[1]), fminf(c[2], c[3])),
                         fminf(fminf(c[4], c[5]), fminf(c[6], c[7])));
        if (!__any(mn < kRC2)) continue;

        float4 pc = PAR4[jcur];                 // one b128 load: (s, t, rad, cadL)
        const float colS = pc.x, colT = pc.y, colRad = pc.z, colCL = pc.w;

#pragma unroll
        for (int v = 0; v < 8; ++v) {
            const int i = rowBase + v + 8 * half;
            float d2 = c[v];
            float r  = __builtin_amdgcn_sqrtf(d2 > 0.0f ? d2 : 1.0f); // raw v_sqrt_f32
            // smooth multiplicative cutoff
            float r2    = r * r;
            float t1    = kRC2 - r2;
            float inner = t1 * t1 * (kRC2 + 2.0f * r2 - 3.0f * kRO2) * kInvDen;
            float f     = (r < kRO) ? 1.0f : ((r < kRC) ? inner : 0.0f);
            if (f != 0.0f && i != jcur) {
                float sig  = rowRad[v] + colRad;
                float e    = __expf(-kAlpha * (r - sig));
                float eps0 = (rowT[v] == colT) ? (rowS[v] + colS) : (2.0f * colCL);
                float eps  = rowAl[v] * (kEpsMax * eps0 + kEpsMin);
                acc += eps * (e * e - 2.0f * e) * f;
            }
        }
    }

    // --- wave32 reduction, one atomic per wave; apply the global 0.5 factor
#pragma unroll
    for (int off = 16; off > 0; off >>= 1) acc += __shfl_xor(acc, off, 32);
    if (lane == 0) atomicAdd(out, 0.5f * acc);
}

extern "C" void kernel_launch(void* const* d_in, const int* in_sizes, int n_in,
                              void* d_out, int out_size, void* d_ws, size_t ws_size,
                              hipStream_t stream) {
    const float* pos = (const float*)d_in[0];   // [N,3]
    const float* ct  = (const float*)d_in[1];   // [N,4] one-hot
    const float* cad = (const float*)d_in[2];   // [N,5]
    const float* rad = (const float*)d_in[3];   // [N,1]
    float* out = (float*)d_out;                 // scalar energy
    float* W   = (float*)d_ws;                  // 9*N floats (221 KB)

    morse_prep_kernel<<<(NPART + 255) / 256, 256, 0, stream>>>(pos, ct, cad, rad, W, out);

    dim3 grid(ROW_BLOCKS, COL_CHUNKS);          // 48 x 16 blocks, 8 waves each
    morse_energy_kernel<<<grid, 256, 0, stream>>>(W, out);
}